// VGAE_Encoder_16569983828163
// MI455X (gfx1250) — compile-verified
//
#include <hip/hip_runtime.h>
#include <hip/hip_bf16.h>

typedef __attribute__((ext_vector_type(2))) float v2f;
typedef __attribute__((ext_vector_type(8))) float v8f;

#define ATOMIC_ADD_F32(p, v) \
  __hip_atomic_fetch_add((p), (v), __ATOMIC_RELAXED, __HIP_MEMORY_SCOPE_AGENT)

// ---------------------------------------------------------------- degree prep
__global__ void k_deg_init(float* __restrict__ deg, int n) {
  int i = blockIdx.x * blockDim.x + threadIdx.x;
  if (i < n) deg[i] = 1.0f;  // self-loop contributes 1
}

__global__ void k_deg_count(const int* __restrict__ dst, float* __restrict__ deg, int e) {
  int i = blockIdx.x * blockDim.x + threadIdx.x;
  if (i < e) ATOMIC_ADD_F32(&deg[dst[i]], 1.0f);
}

__global__ void k_rsqrt_inplace(float* __restrict__ deg, int n) {
  int i = blockIdx.x * blockDim.x + threadIdx.x;
  if (i < n) deg[i] = rsqrtf(deg[i]);  // deg >= 1 always (self-loop)
}

// ---------------------------------------------------------------- WMMA GEMM
// C[nrows, 128] = A[nrows, K] * B[K, 128], fp32 via V_WMMA_F32_16X16X4_F32.
// grid.x = ceil(nrows/16); block = 256 (8 waves); wave w handles cols [16w,16w+16).
// A k-slices (16x32) are staged in LDS once per block and shared by all 8 waves.
#define TK 32
#define APAD 34  // row stride in floats: conflict-free banks + 8B alignment

__global__ void k_gemm_wmma(const float* __restrict__ A, const float* __restrict__ B,
                            float* __restrict__ C, int nrows, int K) {
  __shared__ float As[16 * APAD];

  const int lane = threadIdx.x & 31;
  const int wave = threadIdx.x >> 5;
  const int row0 = blockIdx.x << 4;
  const int col0 = wave << 4;
  const int t  = lane & 15;          // row (A) / col (B) within tile
  const int kk = (lane >> 4) << 1;   // lanes 0-15 -> k,k+1 ; lanes 16-31 -> k+2,k+3

  // cooperative A staging: 16 rows x 32 k = 512 floats, 2 consecutive per thread
  const int lr = threadIdx.x >> 4;         // 0..15 row
  const int lc = (threadIdx.x & 15) << 1;  // 0,2,...,30 col pair
  int srow = row0 + lr;
  if (srow >= nrows) srow = nrows - 1;     // clamp (keeps EXEC full for WMMA)
  const float* aload = A + (size_t)srow * (size_t)K + lc;

  const float* bp = B + (col0 + t);
  v8f acc = {};

  for (int k0 = 0; k0 < K; k0 += TK) {
    __syncthreads();
    As[lr * APAD + lc]     = aload[k0];
    As[lr * APAD + lc + 1] = aload[k0 + 1];
    if (k0 + TK < K) __builtin_prefetch(aload + k0 + TK, 0, 3);
    __syncthreads();

#pragma unroll
    for (int kd = 0; kd < TK; kd += 4) {
      v2f a, b;
      a.x = As[t * APAD + kd + kk];
      a.y = As[t * APAD + kd + kk + 1];
      b.x = bp[(size_t)(k0 + kd + kk) * 128];
      b.y = bp[(size_t)(k0 + kd + kk + 1) * 128];
      acc = __builtin_amdgcn_wmma_f32_16x16x4_f32(
          /*neg_a=*/false, a, /*neg_b=*/false, b,
          /*c_mod=*/(short)0, acc, /*reuse_a=*/false, /*reuse_b=*/false);
    }
  }

  // C/D layout: VGPR i -> row (i or i+8 by lane half), col = lane&15
  const int mb = row0 + ((lane >> 4) << 3);
#pragma unroll
  for (int i = 0; i < 8; ++i) {
    int r = mb + i;
    if (r < nrows) C[(size_t)r * 128 + col0 + t] = acc[i];
  }
}

// Pack Wm[128,64] | Wv[128,64] -> Wmv[128,128]
__global__ void k_pack_w(const float* __restrict__ Wm, const float* __restrict__ Wv,
                         float* __restrict__ Wmv) {
  int idx = blockIdx.x * blockDim.x + threadIdx.x;
  if (idx >= 128 * 128) return;
  int k = idx >> 7, c = idx & 127;
  Wmv[idx] = (c < 64) ? Wm[k * 64 + c] : Wv[k * 64 + (c - 64)];
}

// ---------------------------------------------------------------- aggregation
// h[i,:] = htmp[i,:] * dis[i]^2   (self-loop term seeds the accumulator)
__global__ void k_self_seed(const float* __restrict__ htmp, const float* __restrict__ dis,
                            float* __restrict__ h, int n) {
  int idx = blockIdx.x * blockDim.x + threadIdx.x;
  if (idx >= n * 128) return;
  float d = dis[idx >> 7];
  h[idx] = htmp[idx] * d * d;
}

// One wave per edge: gather htmp[src,:]*norm, atomic-add into h[dst,:]
__global__ void k_edges_h(const int* __restrict__ src, const int* __restrict__ dst,
                          const float* __restrict__ dis, const float* __restrict__ htmp,
                          float* __restrict__ h, int nedges) {
  int e = (blockIdx.x << 3) + (threadIdx.x >> 5);
  if (e >= nedges) return;
  int lane = threadIdx.x & 31;
  int s = src[e], d = dst[e];
  float nrm = dis[s] * dis[d];
  const float4 v = ((const float4*)(htmp + (size_t)s * 128))[lane];
  float* hd = h + (size_t)d * 128 + (lane << 2);
  ATOMIC_ADD_F32(hd + 0, v.x * nrm);
  ATOMIC_ADD_F32(hd + 1, v.y * nrm);
  ATOMIC_ADD_F32(hd + 2, v.z * nrm);
  ATOMIC_ADD_F32(hd + 3, v.w * nrm);
}

__global__ void k_bias_relu(float* __restrict__ h, const float* __restrict__ b, int n) {
  int idx = blockIdx.x * blockDim.x + threadIdx.x;
  if (idx >= n * 128) return;
  h[idx] = fmaxf(h[idx] + b[idx & 127], 0.0f);
}

// Seed d_out with self-loop term of mv (cols 0..63 -> mean, 64..127 -> var)
__global__ void k_self_seed_out(const float* __restrict__ mv, const float* __restrict__ dis,
                                float* __restrict__ out, int n) {
  int idx = blockIdx.x * blockDim.x + threadIdx.x;
  if (idx >= n * 128) return;
  int node = idx >> 7, c = idx & 127;
  float d = dis[node];
  float val = mv[idx] * d * d;
  if (c < 64) out[(size_t)node * 64 + c] = val;
  else        out[(size_t)n * 64 + (size_t)node * 64 + (c - 64)] = val;
}

__global__ void k_edges_out(const int* __restrict__ src, const int* __restrict__ dst,
                            const float* __restrict__ dis, const float* __restrict__ mv,
                            float* __restrict__ out, int nedges, int n) {
  int e = (blockIdx.x << 3) + (threadIdx.x >> 5);
  if (e >= nedges) return;
  int lane = threadIdx.x & 31;
  int s = src[e], d = dst[e];
  float nrm = dis[s] * dis[d];
  const float4 v = ((const float4*)(mv + (size_t)s * 128))[lane];
  float* base = (lane < 16)
      ? out + (size_t)d * 64 + (lane << 2)                                   // mean
      : out + (size_t)n * 64 + (size_t)d * 64 + ((lane - 16) << 2);          // var
  ATOMIC_ADD_F32(base + 0, v.x * nrm);
  ATOMIC_ADD_F32(base + 1, v.y * nrm);
  ATOMIC_ADD_F32(base + 2, v.z * nrm);
  ATOMIC_ADD_F32(base + 3, v.w * nrm);
}

__global__ void k_bias_out(float* __restrict__ out, const float* __restrict__ bm,
                           const float* __restrict__ bv, int n) {
  int idx = blockIdx.x * blockDim.x + threadIdx.x;
  if (idx >= n * 128) return;
  int node = idx >> 7, c = idx & 127;
  if (c < 64) out[(size_t)node * 64 + c] += bm[c];
  else        out[(size_t)n * 64 + (size_t)node * 64 + (c - 64)] += bv[c - 64];
}

// ---------------------------------------------------------------- launch
extern "C" void kernel_launch(void* const* d_in, const int* in_sizes, int n_in,
                              void* d_out, int out_size, void* d_ws, size_t ws_size,
                              hipStream_t stream) {
  (void)n_in; (void)out_size; (void)ws_size;
  const float* X  = (const float*)d_in[0];
  const int*   ei = (const int*)d_in[1];
  const float* W1 = (const float*)d_in[2];
  const float* b1 = (const float*)d_in[3];
  const float* Wm = (const float*)d_in[4];
  const float* bm = (const float*)d_in[5];
  const float* Wv = (const float*)d_in[6];
  const float* bv = (const float*)d_in[7];

  const int n = in_sizes[0] / 512;   // 100000
  const int E = in_sizes[1] / 2;     // 3200000
  const int* src = ei;
  const int* dst = ei + E;
  float* out = (float*)d_out;

  // Workspace layout (floats): deg/dis [nAl] | Wmv [16384] | htmp/mv [n*128] | h [n*128]
  float* ws = (float*)d_ws;
  size_t nAl = ((size_t)n + 127) & ~(size_t)127;
  float* deg  = ws;                 // becomes deg_inv_sqrt in place
  float* wmv  = ws + nAl;
  float* htmp = wmv + 16384;        // reused as mv for layer 2
  float* h    = htmp + (size_t)n * 128;

  const int tpb = 256;
  const int nel = n * 128;
  dim3 gN((n + tpb - 1) / tpb), gE((E + tpb - 1) / tpb);
  dim3 gEl((nel + tpb - 1) / tpb), gEdge((E + 7) / 8), gTile((n + 15) / 16);

  // degrees -> deg_inv_sqrt
  k_deg_init<<<gN, tpb, 0, stream>>>(deg, n);
  k_deg_count<<<gE, tpb, 0, stream>>>(dst, deg, E);
  k_rsqrt_inplace<<<gN, tpb, 0, stream>>>(deg, n);

  // layer 1: htmp = X @ W1 ; h = aggregate(htmp) ; h = relu(h + b1)
  k_gemm_wmma<<<gTile, 256, 0, stream>>>(X, W1, htmp, n, 512);
  k_self_seed<<<gEl, tpb, 0, stream>>>(htmp, deg, h, n);
  k_edges_h<<<gEdge, 256, 0, stream>>>(src, dst, deg, htmp, h, E);
  k_bias_relu<<<gEl, tpb, 0, stream>>>(h, b1, n);

  // layer 2: mv = h @ [Wm|Wv] ; out = aggregate(mv) + [bm|bv]
  k_pack_w<<<(128 * 128 + tpb - 1) / tpb, tpb, 0, stream>>>(Wm, Wv, wmv);
  k_gemm_wmma<<<gTile, 256, 0, stream>>>(h, wmv, htmp, n, 128);
  k_self_seed_out<<<gEl, tpb, 0, stream>>>(htmp, deg, out, n);
  k_edges_out<<<gEdge, 256, 0, stream>>>(src, dst, deg, htmp, out, E, n);
  k_bias_out<<<gEl, tpb, 0, stream>>>(out, bm, bv, n);
}